// FusedLinearMax_8229157339851
// MI455X (gfx1250) — compile-verified
//
#include <hip/hip_runtime.h>
#include <hip/hip_bf16.h>

typedef __attribute__((ext_vector_type(2))) float v2f;
typedef __attribute__((ext_vector_type(8))) float v8f;

#define B_ 256
#define N_ 512
#define D_ 128
#define H_ 256

// One wave computes TWO 16-wide h-tiles for one batch element b:
// each A fragment (x tile, K=4 slice) is reused by two independent
// V_WMMA_F32_16X16X4_F32 accumulation chains -> load:wmma = 1:2 and
// two interleaved XDL dependency chains. One block (8 waves) = one b,
// so all 16 h-tiles of x[b] (256 KB, L2/WGP$-resident) stay on one WGP.
__global__ __launch_bounds__(256) void FusedLinearMax_kernel(
    const float* __restrict__ x, const float* __restrict__ w,
    const float* __restrict__ bias, float* __restrict__ out)
{
  const int lane = threadIdx.x & 31;
  const int wave = threadIdx.x >> 5;   // 0..7
  const int b    = blockIdx.x;
  const int h0   = wave << 5;          // two h-tiles: [h0, h0+16), [h0+16, h0+32)

  const int row  = lane & 15;          // M (A) / N (B) index
  const int koff = (lane >> 4) << 1;   // hi-lanes hold K+2 (ISA 16x4 f32 layout)

  const float* xb  = x + (size_t)b * (N_ * D_);
  const float* wt0 = w + (size_t)(h0 + row) * D_ + koff;
  const float* wt1 = wt0 + 16 * D_;

  // Preload both B tiles (16 h x 128 d each) -> 2 x 32 k-steps x 2 VGPRs.
  v2f bf0[32], bf1[32];
#pragma unroll
  for (int kk = 0; kk < 32; ++kk) {
    bf0[kk] = *(const v2f*)(wt0 + 4 * kk);
    bf1[kk] = *(const v2f*)(wt1 + 4 * kk);
  }

  float m0 = -__builtin_inff();
  float m1 = -__builtin_inff();

#pragma unroll 2
  for (int nt = 0; nt < N_ / 16; ++nt) {
    const float* xa = xb + (size_t)(nt * 16 + row) * D_ + koff;
    v2f af[32];
#pragma unroll
    for (int kk = 0; kk < 32; ++kk) {
      af[kk] = *(const v2f*)(xa + 4 * kk);   // global_load_b64, 8B aligned
    }
    v8f c0 = {0.f, 0.f, 0.f, 0.f, 0.f, 0.f, 0.f, 0.f};
    v8f c1 = {0.f, 0.f, 0.f, 0.f, 0.f, 0.f, 0.f, 0.f};
#pragma unroll
    for (int kk = 0; kk < 32; ++kk) {
      // Two independent D->C chains sharing one A fragment.
      c0 = __builtin_amdgcn_wmma_f32_16x16x4_f32(
          false, af[kk], false, bf0[kk], (short)0, c0, false, false);
      c1 = __builtin_amdgcn_wmma_f32_16x16x4_f32(
          false, af[kk], false, bf1[kk], (short)0, c1, false, false);
    }
    // C/D layout: VGPR r, lane L -> n-row = r + (L>=16 ? 8 : 0), h-col = L&15.
#pragma unroll
    for (int r = 0; r < 8; ++r) {
      m0 = fmaxf(m0, c0[r]);
      m1 = fmaxf(m1, c1[r]);
    }
  }

  // Merge the two n-row halves held by lane L and lane L^16 (same h column).
  m0 = fmaxf(m0, __shfl_xor(m0, 16, 32));
  m1 = fmaxf(m1, __shfl_xor(m1, 16, 32));

  if (lane < 16) {
    const size_t base = (size_t)b * H_ + h0;
    out[base + lane]      = m0 + bias[h0 + lane];
    out[base + 16 + lane] = m1 + bias[h0 + 16 + lane];
  }
}

extern "C" void kernel_launch(void* const* d_in, const int* in_sizes, int n_in,
                              void* d_out, int out_size, void* d_ws, size_t ws_size,
                              hipStream_t stream) {
  (void)in_sizes; (void)n_in; (void)d_ws; (void)ws_size; (void)out_size;
  const float* x    = (const float*)d_in[0];
  const float* w    = (const float*)d_in[1];
  const float* bias = (const float*)d_in[2];
  float* out        = (float*)d_out;

  dim3 grid(B_);     // one block per batch element
  dim3 block(256);   // 8 waves (wave32), each owning 2 h-tiles
  hipLaunchKernelGGL(FusedLinearMax_kernel, grid, block, 0, stream, x, w, bias, out);
}